// LayerStacks_62483184222247
// MI455X (gfx1250) — compile-verified
//
#include <hip/hip_runtime.h>
#include <hip/hip_bf16.h>

// ---------------------------------------------------------------------------
// LayerStacks (NNUE-style) for MI455X / gfx1250, wave32.
//
// Strategy:
//  K0..K2: bucket rows by ls_indices (count / prefix / scatter) -> perm[]
//  KP    : pre-swizzle combined weights (16 selected-stack rows + 16 l1f rows
//          = 32 x 1024 per stack) into V_WMMA_F32_16X16X4_F32 B-fragment
//          layout, with a K-permutation that makes A loads contiguous.
//  KM    : per (stack, 64-row tile) block, 4 waves, each wave owns a
//          16-row x 32-col C tile. K loop: 64 groups of 16 K-values,
//          8 wmma per group (4 chunks x 2 col-halves). Epilogue transposes
//          acc through LDS and runs the tiny 30->32->1 MLP per row.
// ---------------------------------------------------------------------------

typedef __attribute__((ext_vector_type(2))) float v2f;
typedef __attribute__((ext_vector_type(8))) float v8f;

#define NB      65536
#define NL1     1024
#define NSTACK  8
#define NGRP    64              // 1024 / 16
// workspace layout (bytes)
#define WS_CNT     0            // int[8]
#define WS_CUR     64           // int[8]
#define WS_OFF     128          // int[8]
#define WS_PERM    1024         // int[65536]
#define WS_FRAG    (1024 + 65536*4)   // float[8*64*2*256] = 1 MB

// ---------------- bucketing ----------------

__global__ void k_zero(int* w) {
    if (threadIdx.x < 48) w[threadIdx.x] = 0;   // cnt, cur, off regions
}

__global__ void k_count(const int* __restrict__ lsi, int* __restrict__ cnt) {
    int i = blockIdx.x * blockDim.x + threadIdx.x;
    if (i < NB) atomicAdd(&cnt[lsi[i] & 7], 1);
}

__global__ void k_scan(const int* __restrict__ cnt, int* __restrict__ off,
                       int* __restrict__ cur) {
    if (threadIdx.x == 0) {
        int acc = 0;
        for (int s = 0; s < NSTACK; ++s) {
            off[s] = acc;
            cur[s] = acc;
            acc += cnt[s];
        }
    }
}

__global__ void k_scatter(const int* __restrict__ lsi, int* __restrict__ cur,
                          int* __restrict__ perm) {
    int i = blockIdx.x * blockDim.x + threadIdx.x;
    if (i < NB) {
        int pos = atomicAdd(&cur[lsi[i] & 7], 1);
        perm[pos] = i;
    }
}

// ---------------- weight pre-swizzle into B-fragment layout ----------------
// frag[((s*64+g)*2+h)*256 + lane*8 + e] holds W[n][k] with
//   n = (lane&15) + 16*h            (combined col: 0..15 stack, 16..31 l1f)
//   k = g*16 + (lane<16 ? 0 : 8) + e     (K-permuted within 16-wide groups)
// For logical wmma chunk c (0..3) of group g, B VGPR v in lanes 0-15 carries
// physical k = g*16+2c+v, lanes 16-31 carry k = g*16+8+2c+v; A uses the same
// permutation, so the K-dot is preserved.

__global__ void k_prep(const float* __restrict__ l1_w,
                       const float* __restrict__ l1f_w,
                       float* __restrict__ frag) {
    int tid = blockIdx.x * blockDim.x + threadIdx.x;   // 18 bits
    int e    = tid & 7;
    int lane = (tid >> 3) & 31;
    int h    = (tid >> 8) & 1;
    int g    = (tid >> 9) & 63;
    int s    = tid >> 15;
    int n = (lane & 15) + 16 * h;
    int k = g * 16 + ((lane < 16) ? 0 : 8) + e;
    float w = (n < 16) ? l1_w[(size_t)(s * 16 + n) * NL1 + k]
                       : l1f_w[(size_t)(n - 16) * NL1 + k];
    frag[tid] = w;
}

// ---------------- main fused kernel ----------------

__global__ __launch_bounds__(128, 2)
void k_main(const float* __restrict__ x,
            const int* __restrict__ perm,
            const int* __restrict__ off,
            const int* __restrict__ cnt,
            const float* __restrict__ frag,
            const float* __restrict__ l1_b,
            const float* __restrict__ l1f_b,
            const float* __restrict__ l2_w,
            const float* __restrict__ l2_b,
            const float* __restrict__ out_w,
            const float* __restrict__ out_b,
            float* __restrict__ out) {
    __shared__ float lds[4][16][33];

    const int s    = blockIdx.x >> 10;        // stack
    const int tile = blockIdx.x & 1023;       // 64-row tile within stack
    const int cnt_s = cnt[s];
    if (cnt_s == 0) return;
    if (tile * 64 >= cnt_s) return;           // uniform across block

    const int wv   = threadIdx.x >> 5;
    const int lane = threadIdx.x & 31;
    const int r    = lane & 15;               // row within wave tile
    const int p    = tile * 64 + wv * 16 + r; // position within bucket
    const int pcl  = (p < cnt_s) ? p : (cnt_s - 1);
    const int rrow = perm[off[s] + pcl];      // original batch row

    const int koff = (lane < 16) ? 0 : 8;
    const float4* __restrict__ ax =
        (const float4*)(x + (size_t)rrow * NL1 + koff);
    const float4* __restrict__ bx = (const float4*)frag;

    v8f acc0 = {0.f, 0.f, 0.f, 0.f, 0.f, 0.f, 0.f, 0.f};
    v8f acc1 = {0.f, 0.f, 0.f, 0.f, 0.f, 0.f, 0.f, 0.f};

    const int fb0 = ((s * NGRP) * 2 + 0) * 64 + lane * 2;  // float4 index
    const int fb1 = ((s * NGRP) * 2 + 1) * 64 + lane * 2;

#pragma unroll 2
    for (int g = 0; g < NGRP; ++g) {
        float4 a0 = ax[4 * g + 0];
        float4 a1 = ax[4 * g + 1];
        float4 b00 = bx[fb0 + g * 128 + 0];
        float4 b01 = bx[fb0 + g * 128 + 1];
        float4 b10 = bx[fb1 + g * 128 + 0];
        float4 b11 = bx[fb1 + g * 128 + 1];
        float qa[8], q0[8], q1[8];
        *(float4*)&qa[0] = a0;  *(float4*)&qa[4] = a1;
        *(float4*)&q0[0] = b00; *(float4*)&q0[4] = b01;
        *(float4*)&q1[0] = b10; *(float4*)&q1[4] = b11;
#pragma unroll
        for (int c = 0; c < 4; ++c) {
            v2f A  = {qa[2 * c], qa[2 * c + 1]};
            v2f B0 = {q0[2 * c], q0[2 * c + 1]};
            v2f B1 = {q1[2 * c], q1[2 * c + 1]};
            acc0 = __builtin_amdgcn_wmma_f32_16x16x4_f32(
                false, A, false, B0, (short)0, acc0, false, false);
            acc1 = __builtin_amdgcn_wmma_f32_16x16x4_f32(
                false, A, false, B1, (short)0, acc1, false, false);
        }
    }

    // ---- transpose accumulators to LDS: lds[wv][M][N] ----
    const int mbase = (lane >= 16) ? 8 : 0;
    const int ncol  = lane & 15;
#pragma unroll
    for (int v = 0; v < 8; ++v) {
        lds[wv][v + mbase][ncol]      = acc0[v];
        lds[wv][v + mbase][ncol + 16] = acc1[v];
    }
    __builtin_amdgcn_wave_barrier();

    // ---- per-row epilogue: 2 lanes per row, 16 l2 outputs each ----
    const int jh = lane >> 4;          // which half of the 32 l2 outputs
    float t[15];
    float keep;
    {
        float yc15 = lds[wv][r][15] + l1_b[s * 16 + 15];
        float yf15 = lds[wv][r][31] + l1f_b[15];
        keep = yc15 + yf15;            // l1c_out + l1f_out
#pragma unroll
        for (int j = 0; j < 15; ++j) {
            float yc = lds[wv][r][j]      + l1_b[s * 16 + j];
            float yf = lds[wv][r][16 + j] + l1f_b[j];
            t[j] = yc + yf;
        }
    }
    float l1x[30];
#pragma unroll
    for (int j = 0; j < 15; ++j) {
        float sq = t[j] * t[j] * (127.0f / 128.0f);
        l1x[j]      = fminf(fmaxf(sq, 0.f), 1.f);
        l1x[15 + j] = fminf(fmaxf(t[j], 0.f), 1.f);
    }
    float partial = 0.f;
#pragma unroll 4
    for (int oi = 0; oi < 16; ++oi) {
        int o = jh * 16 + oi;
        const float* wrow = l2_w + (size_t)(s * 32 + o) * 30;
        float a2 = l2_b[s * 32 + o];
#pragma unroll
        for (int k = 0; k < 30; ++k) a2 += l1x[k] * wrow[k];
        float l2x = fminf(fmaxf(a2, 0.f), 1.f);
        partial += l2x * out_w[s * 32 + o];
    }
    float tot = partial + __shfl_xor(partial, 16, 32);

    if (lane < 16 && p < cnt_s) {
        out[rrow] = tot + out_b[s] + keep;
    }
}

// ---------------------------------------------------------------------------

extern "C" void kernel_launch(void* const* d_in, const int* in_sizes, int n_in,
                              void* d_out, int out_size, void* d_ws, size_t ws_size,
                              hipStream_t stream) {
    (void)in_sizes; (void)n_in; (void)out_size; (void)ws_size;
    const float* x      = (const float*)d_in[0];
    const int*   lsi    = (const int*)d_in[1];
    const float* l1_w   = (const float*)d_in[2];
    const float* l1_b   = (const float*)d_in[3];
    const float* l1f_w  = (const float*)d_in[4];
    const float* l1f_b  = (const float*)d_in[5];
    const float* l2_w   = (const float*)d_in[6];
    const float* l2_b   = (const float*)d_in[7];
    const float* out_w  = (const float*)d_in[8];
    const float* out_b  = (const float*)d_in[9];
    float* out = (float*)d_out;

    char* ws   = (char*)d_ws;
    int*  cnt  = (int*)(ws + WS_CNT);
    int*  cur  = (int*)(ws + WS_CUR);
    int*  off  = (int*)(ws + WS_OFF);
    int*  perm = (int*)(ws + WS_PERM);
    float* frag = (float*)(ws + WS_FRAG);

    k_zero<<<1, 64, 0, stream>>>(cnt);
    k_count<<<NB / 256, 256, 0, stream>>>(lsi, cnt);
    k_scan<<<1, 32, 0, stream>>>(cnt, off, cur);
    k_scatter<<<NB / 256, 256, 0, stream>>>(lsi, cur, perm);
    k_prep<<<(NSTACK * NGRP * 2 * 256) / 256, 256, 0, stream>>>(l1_w, l1f_w, frag);
    k_main<<<NSTACK * 1024, 128, 0, stream>>>(x, perm, off, cnt, frag,
                                              l1_b, l1f_b, l2_w, l2_b,
                                              out_w, out_b, out);
}